// GeneratorModel_86990267613478
// MI455X (gfx1250) — compile-verified
//
#include <hip/hip_runtime.h>
#include <hip/hip_bf16.h>
#include <math.h>

// Problem constants (from reference)
#define Bb   8
#define Ss   512
#define Dd   256
#define Hh   128
#define G4H  512     // 4*H
#define T1c  101
#define Td   100     // decoder steps
#define Vv   50000
#define KZ   384     // 3*H (logit input dim)
#define NT   3125    // Vv / 16 exactly
#define LDW1 385     // W1 row stride (3H+1)

typedef float v2f __attribute__((ext_vector_type(2)));
typedef float v8f __attribute__((ext_vector_type(8)));

__device__ __forceinline__ float sigmf(float x) { return 1.f / (1.f + __expf(-x)); }

// ---------------------------------------------------------------------------
// Generic fp32 WMMA GEMM: C[M,N] = A[M,K] @ W.T (+ bias1 + bias2), W is [N,ldw]
// One wave per 16x16 output tile; K-loop in steps of 4 (V_WMMA_F32_16X16X4_F32).
// REQUIRES: M and N multiples of 16, K multiple of 4 (all call sites satisfy).
// Fragment layout (ISA 7.12.2, fp32): both A (16x4) and B (4x16) put K={k,k+1}
// in lanes 0-15 and K={k+2,k+3} in lanes 16-31, VGPR index advancing K within
// the half-wave -> each lane loads a contiguous float2.
// ---------------------------------------------------------------------------
__global__ __launch_bounds__(32) void gemm_wmma_f32(
    const float* __restrict__ A, int lda,
    const float* __restrict__ W, int ldw,
    const float* __restrict__ bias1, const float* __restrict__ bias2,
    float* __restrict__ C, int ldc, int M, int N, int K)
{
  const int n0 = blockIdx.x * 16;
  const int m0 = blockIdx.y * 16;
  const int lane = threadIdx.x;      // 0..31 (wave32)
  const int l15  = lane & 15;
  const int hi2  = (lane >> 4) * 2;  // 0 or 2: this half-wave's K sub-pair
  (void)N; (void)M;

  const float* Arow = A + (size_t)(m0 + l15) * lda + hi2;
  const float* Wrow = W + (size_t)(n0 + l15) * ldw + hi2;

  v8f acc = {};
  if ((ldw & 1) == 0) {
    // 8B-aligned W rows: contiguous float2 loads for both fragments
    #pragma unroll 4
    for (int k = 0; k < K; k += 4) {
      v2f a = *(const v2f*)(Arow + k);
      v2f b = *(const v2f*)(Wrow + k);
      acc = __builtin_amdgcn_wmma_f32_16x16x4_f32(false, a, false, b,
                                                  (short)0, acc, false, false);
    }
  } else {
    // odd ldw (W1, stride 385): B rows only 4B-aligned -> scalar B loads
    #pragma unroll 4
    for (int k = 0; k < K; k += 4) {
      v2f a = *(const v2f*)(Arow + k);
      v2f b; b.x = Wrow[k]; b.y = Wrow[k + 1];
      acc = __builtin_amdgcn_wmma_f32_16x16x4_f32(false, a, false, b,
                                                  (short)0, acc, false, false);
    }
  }

  const int n = n0 + l15;
  float bs = 0.f;
  if (bias1) bs += bias1[n];
  if (bias2) bs += bias2[n];
  const int mbase = m0 + ((lane >> 4) ? 8 : 0);   // D: VGPR r -> M=r / M=8+r
  #pragma unroll
  for (int r = 0; r < 8; ++r)
    C[(size_t)(mbase + r) * ldc + n] = acc[r] + bs;
}

// ---------------------------------------------------------------------------
// Embedding gather: emb[b,s,:], reversed x_rev[b,s,:]=emb[b,ridx], decoder xd[b,t,:]
// ---------------------------------------------------------------------------
__global__ void embed_kernel(const int* __restrict__ text, const int* __restrict__ tlen,
                             const int* __restrict__ summ, const float* __restrict__ wv,
                             float* __restrict__ emb, float* __restrict__ xrev,
                             float* __restrict__ xd)
{
  const int idx = blockIdx.x * blockDim.x + threadIdx.x;
  const int totE = Bb * Ss * Dd;
  if (idx < totE) {
    const int d = idx % Dd, s = (idx / Dd) % Ss, b = idx / (Dd * Ss);
    emb[idx] = wv[(size_t)text[b * Ss + s] * Dd + d];
    const int len = tlen[b];
    const int rs = (s < len) ? (len - 1 - s) : s;
    xrev[idx] = wv[(size_t)text[b * Ss + rs] * Dd + d];
  } else if (idx < totE + Bb * Td * Dd) {
    const int j = idx - totE;
    const int d = j % Dd, tt = (j / Dd) % Td, b = j / (Dd * Td);
    xd[j] = wv[(size_t)summ[b * T1c + tt] * Dd + d];   // summary[:, :-1]
  }
}

// ---------------------------------------------------------------------------
// Both masked LSTM recurrences in ONE launch: block = (dir, batch), loops S steps.
// Input gates pre-computed (x@Wih.T + bih + bhh). h/c live in LDS.
// ---------------------------------------------------------------------------
__global__ __launch_bounds__(512) void lstm_kernel(
    const float* __restrict__ gpf, const float* __restrict__ gpb,
    const float* __restrict__ Whh_f, const float* __restrict__ Whh_b,
    const int* __restrict__ tlen,
    float* __restrict__ outf, float* __restrict__ outr,
    float* __restrict__ dech, float* __restrict__ decc)
{
  __shared__ float sh[Hh], sc[Hh], sg[G4H];
  const int dir = blockIdx.x >> 3;
  const int b   = blockIdx.x & 7;
  const float* gpre = dir ? gpb : gpf;
  const float* Whh  = dir ? Whh_b : Whh_f;
  float* out = dir ? outr : outf;
  const int j = threadIdx.x;                 // gate index 0..511
  if (j < Hh) { sh[j] = 0.f; sc[j] = 0.f; }
  __syncthreads();
  const int len = tlen[b];
  const float* wr = Whh + (size_t)j * Hh;

  for (int t = 0; t < Ss; ++t) {
    float g = gpre[((size_t)b * Ss + t) * G4H + j];
    #pragma unroll 16
    for (int k = 0; k < Hh; ++k) g = fmaf(sh[k], wr[k], g);
    sg[j] = g;
    __syncthreads();
    if (j < Hh) {
      const float ig = sg[j], fg = sg[Hh + j], gg = sg[2 * Hh + j], og = sg[3 * Hh + j];
      const float cne = sigmf(fg) * sc[j] + sigmf(ig) * tanhf(gg);
      const float hne = sigmf(og) * tanhf(cne);
      const bool msk = (t < len);
      if (msk) { sh[j] = hne; sc[j] = cne; }
      out[((size_t)b * Ss + t) * Hh + j] = msk ? hne : 0.f;
    }
    __syncthreads();
  }
  if (dir == 0 && j < Hh) { dech[b * Hh + j] = sh[j]; decc[b * Hh + j] = sc[j]; }
}

// text_states = concat(out_f, gather(out_r, ridx) * mask)  -> (B*S, 256)
__global__ void build_ts_kernel(const float* __restrict__ outf, const float* __restrict__ outr,
                                const int* __restrict__ tlen, float* __restrict__ ts)
{
  const int idx = blockIdx.x * blockDim.x + threadIdx.x;
  if (idx >= Bb * Ss * Hh) return;
  const int d = idx % Hh, s = (idx / Hh) % Ss, b = idx / (Hh * Ss);
  const int len = tlen[b];
  const bool m = s < len;
  const int rs = m ? (len - 1 - s) : s;
  const size_t row = (size_t)b * Ss + s;
  ts[row * Dd + d]      = outf[row * Hh + d];
  ts[row * Dd + Hh + d] = m ? outr[((size_t)b * Ss + rs) * Hh + d] : 0.f;
}

// ---------------------------------------------------------------------------
// Decoder LSTM cell (one block): gates for all 8 batches, h_new/c_new (uncommitted),
// and dec_part = h_new @ W1[:,256:384].T
// ---------------------------------------------------------------------------
__global__ __launch_bounds__(512) void dec_cell_kernel(
    const float* __restrict__ dpre, const float* __restrict__ Wdhh,
    const float* __restrict__ dech, const float* __restrict__ decc,
    const float* __restrict__ W1,
    float* __restrict__ hn, float* __restrict__ cn, float* __restrict__ dpart, int t)
{
  __shared__ float sh[Bb][Hh];
  __shared__ float sg[Bb][G4H];
  __shared__ float shn[Bb][Hh];
  const int j = threadIdx.x;
  for (int idx = j; idx < Bb * Hh; idx += 512) sh[idx >> 7][idx & 127] = dech[idx];
  __syncthreads();

  float g[Bb];
  #pragma unroll
  for (int b = 0; b < Bb; ++b) g[b] = dpre[((size_t)b * Td + t) * G4H + j];
  const float* wr = Wdhh + (size_t)j * Hh;
  for (int k = 0; k < Hh; ++k) {
    const float wk = wr[k];
    #pragma unroll
    for (int b = 0; b < Bb; ++b) g[b] = fmaf(sh[b][k], wk, g[b]);
  }
  #pragma unroll
  for (int b = 0; b < Bb; ++b) sg[b][j] = g[b];
  __syncthreads();

  for (int idx = j; idx < Bb * Hh; idx += 512) {
    const int b = idx >> 7, d = idx & 127;
    const float ig = sg[b][d], fg = sg[b][Hh + d], gg = sg[b][2 * Hh + d], og = sg[b][3 * Hh + d];
    const float cne = sigmf(fg) * decc[idx] + sigmf(ig) * tanhf(gg);
    const float hne = sigmf(og) * tanhf(cne);
    cn[idx] = cne; hn[idx] = hne; shn[b][d] = hne;
  }
  __syncthreads();

  for (int idx = j; idx < Bb * Hh; idx += 512) {
    const int b = idx >> 7, d = idx & 127;
    const float* w1r = W1 + (size_t)d * LDW1 + 256;
    float acc = 0.f;
    #pragma unroll 8
    for (int k = 0; k < Hh; ++k) acc = fmaf(shn[b][k], w1r[k], acc);
    dpart[idx] = acc;
  }
}

// ---------------------------------------------------------------------------
// Attention + softmax + coverage + context + z build + masked h/c commit.
// enc_part already includes b1. One block per batch row, thread = source pos.
// ---------------------------------------------------------------------------
__global__ __launch_bounds__(512) void attn_kernel(
    const float* __restrict__ encp, const float* __restrict__ dpart,
    float* __restrict__ cov,
    const float* __restrict__ W1, const float* __restrict__ W2, const float* __restrict__ b2,
    const float* __restrict__ ts, const float* __restrict__ hn, const float* __restrict__ cn,
    const int* __restrict__ slen,
    float* __restrict__ z, float* __restrict__ dech, float* __restrict__ decc,
    float* __restrict__ lacc, int t)
{
  __shared__ float red[512];
  __shared__ float sattn[512];
  const int b = blockIdx.x;
  const int s = threadIdx.x;

  const float cv = cov[b * Ss + s];
  const float* ep = encp + ((size_t)b * Ss + s) * Hh;
  const float* dp = dpart + b * Hh;
  float score = 0.f;
  #pragma unroll 8
  for (int d = 0; d < Hh; ++d) {
    const float hid = tanhf(ep[d] + dp[d] + cv * W1[d * LDW1 + 384]);
    score = fmaf(hid, W2[d], score);
  }
  score += b2[0];

  // softmax over 512 positions
  red[s] = score; __syncthreads();
  for (int off = 256; off > 0; off >>= 1) { if (s < off) red[s] = fmaxf(red[s], red[s + off]); __syncthreads(); }
  const float mx = red[0]; __syncthreads();
  const float e = __expf(score - mx);
  red[s] = e; __syncthreads();
  for (int off = 256; off > 0; off >>= 1) { if (s < off) red[s] += red[s + off]; __syncthreads(); }
  const float inv = 1.f / red[0]; __syncthreads();
  const float attn = e * inv;
  sattn[s] = attn;

  // coverage loss (pre-update cov) and masked coverage update
  red[s] = fminf(cv, attn); __syncthreads();
  for (int off = 256; off > 0; off >>= 1) { if (s < off) red[s] += red[s + off]; __syncthreads(); }
  const bool valid = slen[b] > (t + 1);
  if (s == 0 && valid) atomicAdd(lacc, red[0]);        // GAMMA = 1.0
  if (valid) cov[b * Ss + s] = cv + attn;
  __syncthreads();

  // context = attn @ text_states ; z = [context, h_new]
  if (s < Dd) {
    const float* tsb = ts + (size_t)b * Ss * Dd + s;
    float ctx = 0.f;
    for (int q = 0; q < Ss; ++q) ctx = fmaf(sattn[q], tsb[(size_t)q * Dd], ctx);
    z[b * KZ + s] = ctx;
  } else if (s < KZ) {
    z[b * KZ + s] = hn[b * Hh + (s - Dd)];
  }
  // masked carry commit for next decode step
  if (s < Hh && valid) { dech[b * Hh + s] = hn[b * Hh + s]; decc[b * Hh + s] = cn[b * Hh + s]; }
}

// ---------------------------------------------------------------------------
// Vocab projection tiles via WMMA fp32, emitting online-softmax partials (m, sum).
// One wave per 16-column tile; K=384 in 96 WMMA steps. z rows 8..15 are zero.
// Wl and z rows are 1536B-aligned -> contiguous float2 fragment loads; K-loop
// unrolled x4 so loads clause ahead of the v_wmma chain.
// Lanes 0-15 of the D fragment hold batch rows 0..7 for this tile's 16 columns.
// ---------------------------------------------------------------------------
__global__ __launch_bounds__(128) void vocab_kernel(
    const float* __restrict__ z, const float* __restrict__ Wl, const float* __restrict__ bl,
    float* __restrict__ pm, float* __restrict__ ps)
{
  const int wave = threadIdx.x >> 5;
  const int lane = threadIdx.x & 31;
  const int tile = blockIdx.x * 4 + wave;
  if (tile >= NT) return;
  const int n0 = tile * 16;
  const int l15 = lane & 15;
  const int hi2 = (lane >> 4) * 2;

  const float* zr = z + (size_t)l15 * KZ + hi2;           // A row M=l15
  const float* wr = Wl + (size_t)(n0 + l15) * KZ + hi2;   // B col n0+l15 (L2-resident)
  v8f acc = {};
  #pragma unroll 4
  for (int k = 0; k < KZ; k += 4) {
    v2f a = *(const v2f*)(zr + k);
    v2f b = *(const v2f*)(wr + k);
    acc = __builtin_amdgcn_wmma_f32_16x16x4_f32(false, a, false, b,
                                                (short)0, acc, false, false);
  }

  const float bias = (lane < 16) ? bl[n0 + lane] : 0.f;
  #pragma unroll
  for (int r = 0; r < 8; ++r) {                 // r = batch row (lanes 0-15)
    const float x = acc[r] + bias;
    float m = x;
    for (int w = 1; w < 16; w <<= 1) m = fmaxf(m, __shfl_xor(m, w, 16));
    float ssum = __expf(x - m);
    for (int w = 1; w < 16; w <<= 1) ssum += __shfl_xor(ssum, w, 16);
    if (lane == 0) { pm[(size_t)tile * 8 + r] = m; ps[(size_t)tile * 8 + r] = ssum; }
  }
}

// Combine per-tile partials -> logsumexp; target logit by direct dot; masked NLL.
__global__ __launch_bounds__(256) void lse_kernel(
    const float* __restrict__ pm, const float* __restrict__ ps,
    const float* __restrict__ z, const float* __restrict__ Wl, const float* __restrict__ bl,
    const int* __restrict__ summ, const int* __restrict__ slen,
    float* __restrict__ lacc, int t)
{
  __shared__ float r0[256];
  const int b = blockIdx.x, tid = threadIdx.x;
  float m = -3.4e38f;
  for (int i = tid; i < NT; i += 256) m = fmaxf(m, pm[(size_t)i * 8 + b]);
  r0[tid] = m; __syncthreads();
  for (int off = 128; off > 0; off >>= 1) { if (tid < off) r0[tid] = fmaxf(r0[tid], r0[tid + off]); __syncthreads(); }
  const float M = r0[0]; __syncthreads();
  float ssum = 0.f;
  for (int i = tid; i < NT; i += 256) ssum += ps[(size_t)i * 8 + b] * __expf(pm[(size_t)i * 8 + b] - M);
  r0[tid] = ssum; __syncthreads();
  for (int off = 128; off > 0; off >>= 1) { if (tid < off) r0[tid] += r0[tid + off]; __syncthreads(); }
  const float lse = M + __logf(r0[0]); __syncthreads();

  const int tgt = summ[b * T1c + t + 1];
  float part = 0.f;
  for (int k = tid; k < KZ; k += 256) part = fmaf(z[b * KZ + k], Wl[(size_t)tgt * KZ + k], part);
  r0[tid] = part; __syncthreads();
  for (int off = 128; off > 0; off >>= 1) { if (tid < off) r0[tid] += r0[tid + off]; __syncthreads(); }
  if (tid == 0 && slen[b] > t + 1) atomicAdd(lacc, lse - (r0[0] + bl[tgt]));
}

__global__ void init_kernel(float* __restrict__ cov, float* __restrict__ z, float* __restrict__ lacc)
{
  const int idx = blockIdx.x * blockDim.x + threadIdx.x;
  if (idx < Bb * Ss) cov[idx] = 0.f;
  if (idx < 16 * KZ) z[idx] = 0.f;      // rows 8..15 must stay zero for WMMA padding
  if (idx == 0) lacc[0] = 0.f;
}

__global__ void finalize_kernel(const float* __restrict__ lacc, float* __restrict__ out)
{
  if (threadIdx.x == 0 && blockIdx.x == 0) out[0] = lacc[0] / (float)Td;
}

// ---------------------------------------------------------------------------
extern "C" void kernel_launch(void* const* d_in, const int* in_sizes, int n_in,
                              void* d_out, int out_size, void* d_ws, size_t ws_size,
                              hipStream_t stream)
{
  (void)in_sizes; (void)n_in; (void)out_size; (void)ws_size;
  const int*   text  = (const int*)d_in[0];
  const int*   tlen  = (const int*)d_in[1];
  const int*   summ  = (const int*)d_in[2];
  const int*   slen  = (const int*)d_in[3];
  const float* wv    = (const float*)d_in[4];
  const float* Wih_f = (const float*)d_in[5];
  const float* Whh_f = (const float*)d_in[6];
  const float* bih_f = (const float*)d_in[7];
  const float* bhh_f = (const float*)d_in[8];
  const float* Wih_b = (const float*)d_in[9];
  const float* Whh_b = (const float*)d_in[10];
  const float* bih_b = (const float*)d_in[11];
  const float* bhh_b = (const float*)d_in[12];
  const float* Wdih  = (const float*)d_in[13];
  const float* Wdhh  = (const float*)d_in[14];
  const float* bdih  = (const float*)d_in[15];
  const float* bdhh  = (const float*)d_in[16];
  const float* W1    = (const float*)d_in[17];
  const float* b1    = (const float*)d_in[18];
  const float* W2    = (const float*)d_in[19];
  const float* b2    = (const float*)d_in[20];
  const float* Wl    = (const float*)d_in[21];
  const float* bl    = (const float*)d_in[22];

  float* w = (float*)d_ws;
  auto take = [&](size_t n) { float* p = w; w += n; return p; };
  float* emb  = take((size_t)Bb * Ss * Dd);     // 4096x256
  float* xrev = take((size_t)Bb * Ss * Dd);
  float* xd   = take((size_t)Bb * Td * Dd);     // 800x256
  float* gpf  = take((size_t)Bb * Ss * G4H);    // 4096x512 gate pre-activations
  float* gpb  = take((size_t)Bb * Ss * G4H);
  float* dpre = take((size_t)Bb * Td * G4H);    // 800x512
  float* outf = take((size_t)Bb * Ss * Hh);
  float* outr = take((size_t)Bb * Ss * Hh);
  float* ts   = take((size_t)Bb * Ss * Dd);     // text_states 4096x256
  float* encp = take((size_t)Bb * Ss * Hh);     // enc attention part (+b1) 4096x128
  float* cov  = take((size_t)Bb * Ss);
  float* dech = take(Bb * Hh);
  float* decc = take(Bb * Hh);
  float* hn   = take(Bb * Hh);
  float* cn   = take(Bb * Hh);
  float* dprt = take(Bb * Hh);
  float* zbuf = take(16 * KZ);                  // 16x384, rows 8..15 zero
  float* pm   = take((size_t)NT * 8);
  float* ps   = take((size_t)NT * 8);
  float* lacc = take(1);

  init_kernel<<<(16 * KZ + 255) / 256, 256, 0, stream>>>(cov, zbuf, lacc);

  const int totEmb = Bb * Ss * Dd + Bb * Td * Dd;
  embed_kernel<<<(totEmb + 255) / 256, 256, 0, stream>>>(text, tlen, summ, wv, emb, xrev, xd);

  // Hoisted input-projection GEMMs (biases folded in): gates_pre = x@Wih.T + bih + bhh
  gemm_wmma_f32<<<dim3(G4H / 16, (Bb * Ss) / 16), 32, 0, stream>>>(
      emb, Dd, Wih_f, Dd, bih_f, bhh_f, gpf, G4H, Bb * Ss, G4H, Dd);
  gemm_wmma_f32<<<dim3(G4H / 16, (Bb * Ss) / 16), 32, 0, stream>>>(
      xrev, Dd, Wih_b, Dd, bih_b, bhh_b, gpb, G4H, Bb * Ss, G4H, Dd);
  gemm_wmma_f32<<<dim3(G4H / 16, (Bb * Td) / 16), 32, 0, stream>>>(
      xd, Dd, Wdih, Dd, bdih, bdhh, dpre, G4H, Bb * Td, G4H, Dd);

  // Both LSTM recurrences (fwd+bwd x 8 batches) in one launch
  lstm_kernel<<<16, 512, 0, stream>>>(gpf, gpb, Whh_f, Whh_b, tlen, outf, outr, dech, decc);

  build_ts_kernel<<<(Bb * Ss * Hh + 255) / 256, 256, 0, stream>>>(outf, outr, tlen, ts);

  // Bahdanau precompute: enc_part = text_states @ W1[:, :256].T + b1
  gemm_wmma_f32<<<dim3(Hh / 16, (Bb * Ss) / 16), 32, 0, stream>>>(
      ts, Dd, W1, LDW1, b1, nullptr, encp, Hh, Bb * Ss, Hh, Dd);

  for (int t = 0; t < Td; ++t) {
    dec_cell_kernel<<<1, 512, 0, stream>>>(dpre, Wdhh, dech, decc, W1, hn, cn, dprt, t);
    attn_kernel<<<Bb, 512, 0, stream>>>(encp, dprt, cov, W1, W2, b2, ts, hn, cn, slen,
                                        zbuf, dech, decc, lacc, t);
    vocab_kernel<<<(NT + 3) / 4, 128, 0, stream>>>(zbuf, Wl, bl, pm, ps);
    lse_kernel<<<Bb, 256, 0, stream>>>(pm, ps, zbuf, Wl, bl, summ, slen, lacc, t);
  }

  finalize_kernel<<<1, 32, 0, stream>>>(lacc, (float*)d_out);
}